// SlidingWindowAttention_84705345012270
// MI455X (gfx1250) — compile-verified
//
#include <hip/hip_runtime.h>
#include <math.h>

typedef float v2f __attribute__((ext_vector_type(2)));
typedef float v4f __attribute__((ext_vector_type(4)));
typedef float v8f __attribute__((ext_vector_type(8)));

#define S_LEN 4096
#define D_DIM 64
#define H_NUM 16
#define WIN   256
#define SC_STRIDE 536   // >= 528 columns, odd-ish stride to spread LDS banks

// ---------------------------------------------------------------------------
// Kernel 1: stream zeros over the dense attention matrix (out-of-band part of
// the reference output is exactly 0.0f because exp(-1e9 - max) underflows).
// Nontemporal 16B stores: 1 GiB stream, keep it out of L2 as much as possible.
// ---------------------------------------------------------------------------
__global__ void swa_zero_fill(float* __restrict__ p, size_t n4) {
    size_t i = (size_t)blockIdx.x * blockDim.x + threadIdx.x;
    size_t stride = (size_t)gridDim.x * blockDim.x;
    v4f z = {0.f, 0.f, 0.f, 0.f};
    v4f* p4 = (v4f*)p;
    for (; i < n4; i += stride)
        __builtin_nontemporal_store(z, p4 + i);
}

// ---------------------------------------------------------------------------
// Kernel 2: one wave32 per 16-row query tile. fp32 WMMA (16x16x4) for QK^T
// and P*V; LDS-staged banded scores; two-pass softmax.
// ---------------------------------------------------------------------------
__global__ __launch_bounds__(32)
void swa_attn_kernel(const float* __restrict__ Q, const float* __restrict__ K,
                     const float* __restrict__ V, float* __restrict__ Out,
                     float* __restrict__ Attn)
{
    __shared__ float sc[16][SC_STRIDE];
    __shared__ float rowmax[16];
    __shared__ float rowsum[16];

    const int lane = threadIdx.x;   // 0..31
    const int half = lane >> 4;     // which 16-lane half
    const int l16  = lane & 15;

    const int tile = blockIdx.x & 255;   // 4096/16 = 256 q-tiles per head
    const int head = blockIdx.x >> 8;    // 16 heads
    const int q0   = tile * 16;

    const size_t hb = (size_t)head * S_LEN * D_DIM;
    const float* Qh = Q + hb;
    const float* Kh = K + hb;
    const float* Vh = V + hb;

    int jlo = q0 - WIN;      if (jlo < 0) jlo = 0;
    int jhi = q0 + 15 + WIN; if (jhi > S_LEN - 1) jhi = S_LEN - 1;
    const int ncols  = jhi - jlo + 1;   // multiple of 16, <= 528
    const int ntiles = ncols >> 4;

    // ---- Preload Q fragments (A-matrix, 16x4 fp32 layout):
    //      lanes 0-15 hold M=lane with K={4s, 4s+1}; lanes 16-31 K={4s+2, 4s+3}
    v2f aq[16];
    {
        const float* qrow = Qh + (size_t)(q0 + l16) * D_DIM;
        #pragma unroll
        for (int s = 0; s < 16; ++s) {
            int d0 = 4 * s + 2 * half;
            aq[s].x = qrow[d0];
            aq[s].y = qrow[d0 + 1];
        }
    }

    // ---- Phase 1: banded scores S = (Q K^T) / 8, masked, into LDS ----------
    for (int jt = 0; jt < ntiles; ++jt) {
        const int jb = jlo + jt * 16;
        // B-matrix (64x16 as 16 K-slices of 4): lane l16 holds column n=l16,
        // i.e. Key row (jb+l16); halves hold K offsets {0,1} vs {2,3}.
        const float* krow = Kh + (size_t)(jb + l16) * D_DIM;
        v8f c = {0.f, 0.f, 0.f, 0.f, 0.f, 0.f, 0.f, 0.f};
        #pragma unroll
        for (int s = 0; s < 16; ++s) {
            int d0 = 4 * s + 2 * half;
            v2f b;
            b.x = krow[d0];
            b.y = krow[d0 + 1];
            c = __builtin_amdgcn_wmma_f32_16x16x4_f32(
                    false, aq[s], false, b, (short)0, c, false, false);
        }
        // C/D layout: VGPR r -> (M = r + 8*half, N = l16)
        #pragma unroll
        for (int r = 0; r < 8; ++r) {
            int m   = r + half * 8;
            int row = q0 + m;
            int col = jb + l16;
            float v = c[r] * 0.125f;            // 1/sqrt(64)
            int dd = row - col; dd = dd < 0 ? -dd : dd;
            if (dd > WIN) v = -1e9f;
            sc[m][jt * 16 + l16] = v;
        }
    }
    __syncthreads();

    // ---- Phase 2a: row max (halves stride-2 over columns, combine by shfl) -
    {
        const int row = l16;
        float m = -3.0e38f;
        for (int c0 = half; c0 < ncols; c0 += 2)
            m = fmaxf(m, sc[row][c0]);
        m = fmaxf(m, __shfl_xor(m, 16, 32));
        if (half == 0) rowmax[row] = m;
    }
    __syncthreads();

    // ---- Phase 2b: exp + row sum, store exp back to LDS --------------------
    {
        const int row = l16;
        const float mx = rowmax[row];
        float s = 0.f;
        for (int c0 = half; c0 < ncols; c0 += 2) {
            float e = expf(sc[row][c0] - mx);
            sc[row][c0] = e;
            s += e;
        }
        s += __shfl_xor(s, 16, 32);
        if (half == 0) rowsum[row] = s;
    }
    __syncthreads();

    // ---- Phase 2c: normalize, keep in LDS, stream in-band attn to global ---
    for (int row = 0; row < 16; ++row) {
        const float rs = 1.0f / rowsum[row];
        float* dst = Attn + ((size_t)head * S_LEN + q0 + row) * S_LEN + jlo;
        for (int c0 = lane; c0 < ncols; c0 += 32) {
            float p = sc[row][c0] * rs;
            sc[row][c0] = p;
            __builtin_nontemporal_store(p, dst + c0);
        }
    }
    __syncthreads();

    // ---- Phase 3: Out = P * V via fp32 WMMA --------------------------------
    #pragma unroll
    for (int db = 0; db < 4; ++db) {
        const int dbase = db * 16;
        v8f acc = {0.f, 0.f, 0.f, 0.f, 0.f, 0.f, 0.f, 0.f};
        for (int jt = 0; jt < ntiles; ++jt) {
            const int cb = jt * 16;
            const int jb = jlo + cb;
            #pragma unroll
            for (int kk = 0; kk < 16; kk += 4) {
                const int ko = kk + 2 * half;
                // A = probability tile from LDS (M=l16, K slice per half)
                v2f a;
                a.x = sc[l16][cb + ko];
                a.y = sc[l16][cb + ko + 1];
                // B = V tile: lane column n=l16 is feature (dbase+l16)
                v2f b;
                b.x = Vh[(size_t)(jb + ko)     * D_DIM + dbase + l16];
                b.y = Vh[(size_t)(jb + ko + 1) * D_DIM + dbase + l16];
                acc = __builtin_amdgcn_wmma_f32_16x16x4_f32(
                        false, a, false, b, (short)0, acc, false, false);
            }
        }
        #pragma unroll
        for (int r = 0; r < 8; ++r) {
            int m = r + half * 8;
            Out[((size_t)head * S_LEN + q0 + m) * D_DIM + dbase + l16] = acc[r];
        }
    }
}

extern "C" void kernel_launch(void* const* d_in, const int* in_sizes, int n_in,
                              void* d_out, int out_size, void* d_ws, size_t ws_size,
                              hipStream_t stream) {
    (void)in_sizes; (void)n_in; (void)out_size; (void)d_ws; (void)ws_size;

    const float* Q = (const float*)d_in[0];
    const float* K = (const float*)d_in[1];
    const float* V = (const float*)d_in[2];

    float* Out  = (float*)d_out;
    float* Attn = Out + (size_t)H_NUM * S_LEN * D_DIM;   // tuple: (out, attn)

    // 1) zero-fill the dense attention matrix (out-of-band entries are 0.0f)
    const size_t attn_elems = (size_t)H_NUM * S_LEN * S_LEN;
    swa_zero_fill<<<4096, 256, 0, stream>>>(Attn, attn_elems / 4);

    // 2) banded attention: one wave32 per 16-row query tile
    dim3 grid(H_NUM * (S_LEN / 16));
    swa_attn_kernel<<<grid, 32, 0, stream>>>(Q, K, V, Out, Attn);
}